// GeometricPointNet_5007931867763
// MI455X (gfx1250) — compile-verified
//
#include <hip/hip_runtime.h>
#include <cstdint>
#include <cstddef>

#define N_NODESC 100000
#define N_EDGESC 3200000
#define N_GRAPHS 8
#define HIDDENC 16
#define DEPTHC 10
#define LF_W 168           // 167 local-feature cols padded to multiple of 4
#define EPSC 1e-5f
#define KCHUNK 32

typedef __attribute__((ext_vector_type(2))) float v2f;
typedef __attribute__((ext_vector_type(8))) float v8f;
typedef __attribute__((ext_vector_type(4))) unsigned v4u;
typedef __attribute__((ext_vector_type(8))) int v8i_t;
typedef __attribute__((ext_vector_type(4))) int v4i_t;

// ---------------- utility kernels ----------------
__global__ void k_zero_f32(float* p, size_t n) {
  size_t i = (size_t)blockIdx.x * blockDim.x + threadIdx.x;
  if (i < n) p[i] = 0.f;
}

__global__ void k_fill_u32(unsigned* p, unsigned v, size_t n) {
  size_t i = (size_t)blockIdx.x * blockDim.x + threadIdx.x;
  if (i < n) p[i] = v;
}

// Build padded x_sage [N,8] and write cols 160..167 of local_features slab
__global__ void k_build_inputs(const float* __restrict__ pos, const float* __restrict__ nrm,
                               const float* __restrict__ cur, float* __restrict__ xs,
                               float* __restrict__ lf) {
  int i = blockIdx.x * blockDim.x + threadIdx.x;
  if (i >= N_NODESC) return;
  float v[8];
  v[0] = pos[i * 3 + 0]; v[1] = pos[i * 3 + 1]; v[2] = pos[i * 3 + 2];
  v[3] = nrm[i * 3 + 0]; v[4] = nrm[i * 3 + 1]; v[5] = nrm[i * 3 + 2];
  v[6] = cur[i];         v[7] = 0.f;
#pragma unroll
  for (int c = 0; c < 8; ++c) {
    xs[(size_t)i * 8 + c] = v[c];
    lf[(size_t)i * LF_W + 160 + c] = v[c];
  }
}

// Zero-pad a [rows,C] weight to [padrows,C]
__global__ void k_padw(const float* __restrict__ w, float* __restrict__ wp,
                       int rows, int padrows, int C) {
  int idx = blockIdx.x * blockDim.x + threadIdx.x;
  if (idx >= padrows * C) return;
  int r = idx / C, c = idx - r * C;
  wp[idx] = (r < rows) ? w[(size_t)r * C + c] : 0.f;
}

// ---------------- SAGE aggregation ----------------
__global__ void k_count(const int* __restrict__ dst, float* __restrict__ cnt) {
  int e = blockIdx.x * blockDim.x + threadIdx.x;
  if (e >= N_EDGESC) return;
  atomicAdd(&cnt[dst[e]], 1.f);
}

__global__ void k_scatter(const float* __restrict__ x, int xstride, int cin,
                          const int* __restrict__ src, const int* __restrict__ dst,
                          float* __restrict__ agg) {
  int e = blockIdx.x * blockDim.x + threadIdx.x;
  if (e >= N_EDGESC) return;
  const float* xr = x + (size_t)src[e] * xstride;
  float* ar = agg + (size_t)dst[e] * HIDDENC;
  for (int c = 0; c < cin; ++c) atomicAdd(&ar[c], xr[c]);
}

// h = (agg/max(cnt,1)) @ w_nb + b_nb + x @ w_rt     (Cin x 16 weights, uniform -> scalar loads)
__global__ void k_sage_node(const float* __restrict__ x, int xstride, int cin,
                            const float* __restrict__ agg, const float* __restrict__ cnt,
                            const float* __restrict__ w_nb, const float* __restrict__ b_nb,
                            const float* __restrict__ w_rt, float* __restrict__ h) {
  int i = blockIdx.x * blockDim.x + threadIdx.x;
  if (i >= N_NODESC) return;
  float acc[HIDDENC];
#pragma unroll
  for (int o = 0; o < HIDDENC; ++o) acc[o] = b_nb[o];
  float inv = 1.f / fmaxf(cnt[i], 1.f);
  for (int c = 0; c < cin; ++c) {
    float xa = agg[(size_t)i * HIDDENC + c] * inv;
    float xr = x[(size_t)i * xstride + c];
#pragma unroll
    for (int o = 0; o < HIDDENC; ++o)
      acc[o] += xa * w_nb[c * HIDDENC + o] + xr * w_rt[c * HIDDENC + o];
  }
#pragma unroll
  for (int o = 0; o < HIDDENC; ++o) h[(size_t)i * HIDDENC + o] = acc[o];
}

// ---------------- BatchNorm (training-mode stats) ----------------
__global__ void k_stats(const float* __restrict__ y, int C, float* __restrict__ gsums) {
  __shared__ float s_sum[1024];
  __shared__ float s_ss[1024];
  for (int c = threadIdx.x; c < C; c += blockDim.x) { s_sum[c] = 0.f; s_ss[c] = 0.f; }
  __syncthreads();
  size_t r0 = (size_t)blockIdx.x * 256;
  size_t rend = r0 + 256; if (rend > N_NODESC) rend = N_NODESC;
  size_t begin = r0 * (size_t)C + threadIdx.x, end = rend * (size_t)C;
  unsigned mask = (unsigned)C - 1u;
  for (size_t idx = begin; idx < end; idx += blockDim.x) {
    float v = y[idx];
    int c = (int)((unsigned)idx & mask);
    atomicAdd(&s_sum[c], v);
    atomicAdd(&s_ss[c], v * v);
  }
  __syncthreads();
  for (int c = threadIdx.x; c < C; c += blockDim.x) {
    atomicAdd(&gsums[c], s_sum[c]);
    atomicAdd(&gsums[C + c], s_ss[c]);
  }
}

__global__ void k_finalize(const float* __restrict__ gsums, int C,
                           float* __restrict__ mean, float* __restrict__ var) {
  int c = blockIdx.x * blockDim.x + threadIdx.x;
  if (c >= C) return;
  float m = gsums[c] / (float)N_NODESC;
  mean[c] = m;
  var[c] = gsums[C + c] / (float)N_NODESC - m * m;   // population variance (ddof=0)
}

__global__ void k_bn_elu_sage(const float* __restrict__ h, const float* __restrict__ g,
                              const float* __restrict__ b, const float* __restrict__ mean,
                              const float* __restrict__ var, float* __restrict__ xbuf,
                              float* __restrict__ lf, int layer) {
  size_t idx = (size_t)blockIdx.x * blockDim.x + threadIdx.x;
  if (idx >= (size_t)N_NODESC * HIDDENC) return;
  int c = (int)(idx & 15u);
  size_t i = idx >> 4;
  float z = g[c] * (h[idx] - mean[c]) * rsqrtf(var[c] + EPSC) + b[c];
  float e = z > 0.f ? z : (expf(z) - 1.f);
  xbuf[idx] = e;
  lf[i * LF_W + layer * HIDDENC + c] = e;
}

__global__ void k_bn_apply(float* __restrict__ y, const float* __restrict__ g,
                           const float* __restrict__ b, const float* __restrict__ mean,
                           const float* __restrict__ var, int C, size_t total) {
  size_t idx = (size_t)blockIdx.x * blockDim.x + threadIdx.x;
  if (idx >= total) return;
  int c = (int)((unsigned)idx & ((unsigned)C - 1u));
  y[idx] = g[c] * (y[idx] - mean[c]) * rsqrtf(var[c] + EPSC) + b[c];
}

// ---------------- TDM: DMA one 32x64 f32 W-slab into LDS (row stride 65 via pad fields) ----------------
// D# packing per CDNA5 ISA 8.3-8.5. 2D tile; tensor_dim1 clamped so OOB K-rows zero-fill.
__device__ __forceinline__ void tdm_load_B(const float* __restrict__ W, int K, int C,
                                           int kc, int n0, unsigned lds_byte_addr) {
  unsigned long long ga = (unsigned long long)(uintptr_t)(W + (size_t)kc * C + n0);
  int rows = K - kc; if (rows > KCHUNK) rows = KCHUNK;

  v4u g0;
  g0.x = 1u;                                      // count=1, is_restore=0, gather off
  g0.y = lds_byte_addr;                           // lds_addr (bytes)
  g0.z = (unsigned)(ga & 0xFFFFFFFFu);            // global_addr[31:0]
  g0.w = (unsigned)((ga >> 32) & 0x01FFFFFFu)     // global_addr[56:32]
       | (2u << 30);                              // type = 2 ("image")

  v8i_t g1;
  g1[0] = (int)((2u << 16)                        // data_size = 4 bytes
              | (1u << 20)                        // pad_enable
              | (5u << 22)                        // pad_interval: 64 DWORDs
              | (0u << 25));                      // pad_amount: 1 DWORD  -> stride 65 floats
  g1[1] = (int)(64u << 16);                       // tensor_dim0[15:0]=64 (bits 63:48)
  g1[2] = (int)(((unsigned)rows & 0xFFFFu) << 16);// tensor_dim0 hi=0 | tensor_dim1[15:0]=rows
  g1[3] = (int)(64u << 16);                       // tensor_dim1 hi=0 | tile_dim0=64
  g1[4] = (int)(KCHUNK);                          // tile_dim1=32 | tile_dim2=0
  g1[5] = (int)((unsigned)C);                     // tensor_dim0_stride[31:0] = C elements
  g1[6] = 0;                                      // stride0 hi | tensor_dim1_stride lo (2D: unused)
  g1[7] = 0;

  v4i_t gz4 = {0, 0, 0, 0};                       // groups 2/3: unused for 2D tile
  v8i_t gz8 = {0, 0, 0, 0, 0, 0, 0, 0};           // 6-arg toolchain: extra group, unused
  __builtin_amdgcn_tensor_load_to_lds(g0, g1, gz4, gz4, gz8, 0);
}

// ---------------- f32 WMMA GEMM: out[N,C] = relu?(X[N,K] @ W[K,C] + bias (+ rowBias[batch])) ----------------
// Wave tile 16x64 (4x v8f acc), block = 4 waves = 64 rows x 64 cols.
// B slabs DMA'd by the TDM into double-buffered LDS; compute overlaps the next slab's DMA.
// A layout (ISA 16x4 f32): lanes 0-15 -> K{0,1}, lanes 16-31 -> K{2,3}; M = lane&15.
// C/D layout: col = lane&15, row = vgpr + 8*(lane>>4).
__global__ void __launch_bounds__(128)
k_gemm_wmma(const float* __restrict__ X, const float* __restrict__ W,
            const float* __restrict__ bias, const float* __restrict__ rowBias,
            const int* __restrict__ batch, float* __restrict__ out,
            int K, int C, int relu) {
  __shared__ float Bs[2][KCHUNK][65];
  const int lane = threadIdx.x & 31;
  const int wave = threadIdx.x >> 5;
  const int lm = lane & 15;
  const int lk = (lane >> 4) << 1;            // 0 or 2
  const int n0 = blockIdx.y * 64;
  const int m0 = (blockIdx.x * 4 + wave) * 16;
  const bool active = (m0 < N_NODESC);        // wave-uniform -> EXEC all-ones inside
  v8f acc0 = {}, acc1 = {}, acc2 = {}, acc3 = {};
  const float* xrow = X + (size_t)(m0 + lm) * K;
  const int nChunks = (K + KCHUNK - 1) / KCHUNK;

#define GEMM_STEP(BUF, KK)                                                          \
  {                                                                                 \
    v2f a = *(const v2f*)(xrow + kc + (KK) + lk);  /* 8B aligned: all even */       \
    const int r0 = (KK) + lk, r1 = r0 + 1;                                          \
    v2f b0, b1, b2, b3;                                                             \
    b0.x = Bs[BUF][r0][lm];      b0.y = Bs[BUF][r1][lm];                            \
    b1.x = Bs[BUF][r0][16 + lm]; b1.y = Bs[BUF][r1][16 + lm];                       \
    b2.x = Bs[BUF][r0][32 + lm]; b2.y = Bs[BUF][r1][32 + lm];                       \
    b3.x = Bs[BUF][r0][48 + lm]; b3.y = Bs[BUF][r1][48 + lm];                       \
    acc0 = __builtin_amdgcn_wmma_f32_16x16x4_f32(false, a, false, b0, (short)0, acc0, false, false); \
    acc1 = __builtin_amdgcn_wmma_f32_16x16x4_f32(false, a, false, b1, (short)0, acc1, false, false); \
    acc2 = __builtin_amdgcn_wmma_f32_16x16x4_f32(false, a, false, b2, (short)0, acc2, false, false); \
    acc3 = __builtin_amdgcn_wmma_f32_16x16x4_f32(false, a, false, b3, (short)0, acc3, false, false); \
  }

  if (wave == 0)
    tdm_load_B(W, K, C, 0, n0, (unsigned)(uintptr_t)&Bs[0][0][0]);

  int cur = 0;
  for (int ci = 0; ci < nChunks; ++ci) {
    const int kc = ci * KCHUNK;
    if (wave == 0) {
      if (ci + 1 < nChunks) {
        tdm_load_B(W, K, C, kc + KCHUNK, n0, (unsigned)(uintptr_t)&Bs[cur ^ 1][0][0]);
        __builtin_amdgcn_s_wait_tensorcnt(1);   // chunk ci's DMA complete (in-order)
      } else {
        __builtin_amdgcn_s_wait_tensorcnt(0);
      }
    }
    __syncthreads();                            // publish DMA'd slab to all waves
    const int kcnt = min(KCHUNK, K - kc);       // 32 or 8 (K in {168,256,512})
    if (active) {
      if (kcnt == KCHUNK) {
#pragma unroll
        for (int kk = 0; kk < KCHUNK; kk += 4) GEMM_STEP(cur, kk)
      } else {
        for (int kk = 0; kk < kcnt; kk += 4) GEMM_STEP(cur, kk)
      }
    }
    __syncthreads();                            // slab free before TDM overwrites it
    cur ^= 1;
  }
#undef GEMM_STEP

  if (!active) return;
  const int half = lane >> 4;
#pragma unroll
  for (int j = 0; j < 4; ++j) {
    v8f acc = (j == 0) ? acc0 : (j == 1) ? acc1 : (j == 2) ? acc2 : acc3;
    const int n = n0 + j * 16 + lm;
    const float bb = bias ? bias[n] : 0.f;
#pragma unroll
    for (int v = 0; v < 8; ++v) {
      const int row = m0 + v + 8 * half;
      float val = acc[v] + bb;
      if (rowBias) val += rowBias[(size_t)batch[row] * C + n];
      if (relu) val = fmaxf(val, 0.f);
      out[(size_t)row * C + n] = val;
    }
  }
}

// ---------------- segment-max pooling ----------------
__device__ __forceinline__ unsigned enc_f32(float f) {
  unsigned u = __float_as_uint(f);
  return (u & 0x80000000u) ? ~u : (u | 0x80000000u);
}
__device__ __forceinline__ float dec_f32(unsigned u) {
  u = (u & 0x80000000u) ? (u & 0x7FFFFFFFu) : ~u;
  return __uint_as_float(u);
}
#define ENC_NEG_INF 0x007FFFFFu   // enc(-inf)

__global__ void k_poolmax(const float* __restrict__ g3, const int* __restrict__ batch,
                          unsigned* __restrict__ pooled_u) {
  int c = blockIdx.y * 256 + threadIdx.x;
  size_t r0 = (size_t)blockIdx.x * 512;
  size_t rend = r0 + 512; if (rend > N_NODESC) rend = N_NODESC;
  int curb = -1; float best = 0.f;
  for (size_t r = r0; r < rend; ++r) {
    int b = batch[r];
    float v = g3[r * 1024 + c];
    if (b != curb) {
      if (curb >= 0) atomicMax(&pooled_u[(size_t)curb * 1024 + c], enc_f32(best));
      curb = b; best = v;
    } else best = fmaxf(best, v);
  }
  if (curb >= 0) atomicMax(&pooled_u[(size_t)curb * 1024 + c], enc_f32(best));
}

// pooledW[8,512] = dec(pooled) @ p_w1[167:1191,:] + p_b1
__global__ void k_pooledW(const unsigned* __restrict__ pooled_u, const float* __restrict__ p_w1,
                          const float* __restrict__ p_b1, float* __restrict__ pooledW) {
  int t = blockIdx.x * blockDim.x + threadIdx.x;
  if (t >= N_GRAPHS * 512) return;
  int g = t >> 9, n = t & 511;
  float s = p_b1[n];
  for (int k = 0; k < 1024; ++k)
    s += dec_f32(pooled_u[(size_t)g * 1024 + k]) * p_w1[(size_t)(167 + k) * 512 + n];
  pooledW[t] = s;
}

// final: out[i] = p3[i,:] @ out_w + out_b
__global__ void k_final(const float* __restrict__ p3, const float* __restrict__ out_w,
                        const float* __restrict__ out_b, float* __restrict__ out) {
  int i = blockIdx.x * blockDim.x + threadIdx.x;
  if (i >= N_NODESC) return;
  float s = out_b[0];
  const float* row = p3 + (size_t)i * 128;
#pragma unroll 4
  for (int k = 0; k < 128; ++k) s += row[k] * out_w[k];
  out[i] = s;
}

// ---------------- host orchestration ----------------
static inline unsigned cdiv(size_t a, unsigned b) { return (unsigned)((a + b - 1) / b); }

extern "C" void kernel_launch(void* const* d_in, const int* in_sizes, int n_in,
                              void* d_out, int out_size, void* d_ws, size_t ws_size,
                              hipStream_t stream) {
  (void)in_sizes; (void)n_in; (void)out_size; (void)ws_size;
  const float* pos   = (const float*)d_in[0];
  const float* nrm   = (const float*)d_in[1];
  const float* cur   = (const float*)d_in[2];
  const int*   eidx  = (const int*)d_in[3];
  const int*   batch = (const int*)d_in[4];
  const float* w_nb0 = (const float*)d_in[5];
  const float* b_nb0 = (const float*)d_in[6];
  const float* w_rt0 = (const float*)d_in[7];
  const float* w_nb  = (const float*)d_in[8];
  const float* b_nb  = (const float*)d_in[9];
  const float* w_rt  = (const float*)d_in[10];
  const float* bn_g  = (const float*)d_in[11];
  const float* bn_b  = (const float*)d_in[12];
  const float* g_w1  = (const float*)d_in[13];
  const float* g_b1  = (const float*)d_in[14];
  const float* g_g1  = (const float*)d_in[15];
  const float* g_be1 = (const float*)d_in[16];
  const float* g_w2  = (const float*)d_in[17];
  const float* g_b2  = (const float*)d_in[18];
  const float* g_g2  = (const float*)d_in[19];
  const float* g_be2 = (const float*)d_in[20];
  const float* g_w3  = (const float*)d_in[21];
  const float* g_b3  = (const float*)d_in[22];
  const float* g_g3  = (const float*)d_in[23];
  const float* g_be3 = (const float*)d_in[24];
  const float* p_w1  = (const float*)d_in[25];
  const float* p_b1  = (const float*)d_in[26];
  const float* p_g1  = (const float*)d_in[27];
  const float* p_be1 = (const float*)d_in[28];
  const float* p_w2  = (const float*)d_in[29];
  const float* p_b2  = (const float*)d_in[30];
  const float* p_g2  = (const float*)d_in[31];
  const float* p_be2 = (const float*)d_in[32];
  const float* p_w3  = (const float*)d_in[33];
  const float* p_b3  = (const float*)d_in[34];
  const float* p_g3  = (const float*)d_in[35];
  const float* p_be3 = (const float*)d_in[36];
  const float* out_w = (const float*)d_in[37];
  const float* out_b = (const float*)d_in[38];
  const int* src = eidx;
  const int* dst = eidx + N_EDGESC;
  float* outp = (float*)d_out;

  uint8_t* base = (uint8_t*)d_ws;
  size_t off = 0;
  auto carve = [&](size_t bytes) -> void* {
    void* p = base + off;
    off += (bytes + 255) & ~(size_t)255;
    return p;
  };
  float*    lf       = (float*)carve((size_t)N_NODESC * LF_W * 4);
  float*    xs       = (float*)carve((size_t)N_NODESC * 8 * 4);
  float*    xbuf     = (float*)carve((size_t)N_NODESC * 16 * 4);
  float*    hbuf     = (float*)carve((size_t)N_NODESC * 16 * 4);
  float*    agg      = (float*)carve((size_t)N_NODESC * 16 * 4);
  float*    cnt      = (float*)carve((size_t)N_NODESC * 4);
  float*    g1       = (float*)carve((size_t)N_NODESC * 256 * 4);
  float*    g2       = (float*)carve((size_t)N_NODESC * 512 * 4);
  float*    g3       = (float*)carve((size_t)N_NODESC * 1024 * 4);
  float*    gsums    = (float*)carve(2048 * 4);
  float*    meanb    = (float*)carve(1024 * 4);
  float*    varb     = (float*)carve(1024 * 4);
  unsigned* pooled_u = (unsigned*)carve((size_t)N_GRAPHS * 1024 * 4);
  float*    pooledW  = (float*)carve((size_t)N_GRAPHS * 512 * 4);
  float*    g_w1p    = (float*)carve((size_t)LF_W * 256 * 4);
  float*    p_w1p    = (float*)carve((size_t)LF_W * 512 * 4);
  float* p1 = g2;   // lifetimes don't overlap
  float* p2 = g1;
  float* p3 = g3;

  const unsigned TB = 256;
  const unsigned nodeBlk = cdiv(N_NODESC, TB);
  const unsigned edgeBlk = cdiv(N_EDGESC, TB);
  const unsigned statsBlk = cdiv(N_NODESC, 256);
  const unsigned gemmBx = (N_NODESC / 16 + 3) / 4;   // 1563 blocks of 4 waves

  // --- setup: x_sage, lf tail, edge counts, padded weights ---
  k_build_inputs<<<nodeBlk, TB, 0, stream>>>(pos, nrm, cur, xs, lf);
  k_zero_f32<<<cdiv(N_NODESC, TB), TB, 0, stream>>>(cnt, N_NODESC);
  k_count<<<edgeBlk, TB, 0, stream>>>(dst, cnt);
  k_padw<<<cdiv((size_t)LF_W * 256, TB), TB, 0, stream>>>(g_w1, g_w1p, 167, LF_W, 256);
  k_padw<<<cdiv((size_t)LF_W * 512, TB), TB, 0, stream>>>(p_w1, p_w1p, 167, LF_W, 512);

  // --- SAGE stack ---
  for (int layer = 0; layer < DEPTHC; ++layer) {
    const float* xin = (layer == 0) ? xs : xbuf;
    int xstride = (layer == 0) ? 8 : 16;
    int cin = (layer == 0) ? 7 : 16;
    const float* wnb = (layer == 0) ? w_nb0 : (w_nb + (size_t)(layer - 1) * 256);
    const float* bnb = (layer == 0) ? b_nb0 : (b_nb + (size_t)(layer - 1) * 16);
    const float* wrt = (layer == 0) ? w_rt0 : (w_rt + (size_t)(layer - 1) * 256);
    k_zero_f32<<<cdiv((size_t)N_NODESC * 16, TB), TB, 0, stream>>>(agg, (size_t)N_NODESC * 16);
    k_scatter<<<edgeBlk, TB, 0, stream>>>(xin, xstride, cin, src, dst, agg);
    k_sage_node<<<nodeBlk, TB, 0, stream>>>(xin, xstride, cin, agg, cnt, wnb, bnb, wrt, hbuf);
    k_zero_f32<<<1, 64, 0, stream>>>(gsums, 32);
    k_stats<<<statsBlk, 256, 0, stream>>>(hbuf, 16, gsums);
    k_finalize<<<1, 16, 0, stream>>>(gsums, 16, meanb, varb);
    k_bn_elu_sage<<<cdiv((size_t)N_NODESC * 16, TB), TB, 0, stream>>>(
        hbuf, bn_g + layer * 16, bn_b + layer * 16, meanb, varb, xbuf, lf, layer);
  }

  // --- g-MLP: 167(168) -> 256 -> 512 -> 1024, Lin+ReLU+BN each ---
  struct Stage { const float* X; const float* W; const float* b; const float* g; const float* be;
                 float* Y; int K; int C; };
  const Stage gstages[3] = {
    { lf, g_w1p, g_b1, g_g1, g_be1, g1, LF_W, 256 },
    { g1, g_w2,  g_b2, g_g2, g_be2, g2, 256,  512 },
    { g2, g_w3,  g_b3, g_g3, g_be3, g3, 512,  1024 },
  };
  for (int s = 0; s < 3; ++s) {
    const Stage& st = gstages[s];
    dim3 grid(gemmBx, st.C / 64);
    k_gemm_wmma<<<grid, 128, 0, stream>>>(st.X, st.W, st.b, nullptr, nullptr, st.Y, st.K, st.C, 1);
    k_zero_f32<<<cdiv((size_t)2 * st.C, TB), TB, 0, stream>>>(gsums, (size_t)2 * st.C);
    k_stats<<<statsBlk, 256, 0, stream>>>(st.Y, st.C, gsums);
    k_finalize<<<cdiv(st.C, TB), TB, 0, stream>>>(gsums, st.C, meanb, varb);
    k_bn_apply<<<cdiv((size_t)N_NODESC * st.C, TB), TB, 0, stream>>>(
        st.Y, st.g, st.be, meanb, varb, st.C, (size_t)N_NODESC * st.C);
  }

  // --- segment-max pooling + fold pooled part of p_w1 into an 8x512 per-graph bias table ---
  k_fill_u32<<<cdiv((size_t)N_GRAPHS * 1024, TB), TB, 0, stream>>>(
      pooled_u, ENC_NEG_INF, (size_t)N_GRAPHS * 1024);
  { dim3 grid(cdiv(N_NODESC, 512), 4);
    k_poolmax<<<grid, 256, 0, stream>>>(g3, batch, pooled_u); }
  k_pooledW<<<cdiv((size_t)N_GRAPHS * 512, TB), TB, 0, stream>>>(pooled_u, p_w1, p_b1, pooledW);

  // --- p-MLP: [local|global](1191) -> 512 -> 256 -> 128 ---
  { dim3 grid(gemmBx, 512 / 64);   // p1 = lf @ p_w1[:167] + pooledW[batch]  (bias folded into pooledW)
    k_gemm_wmma<<<grid, 128, 0, stream>>>(lf, p_w1p, nullptr, pooledW, batch, p1, LF_W, 512, 1); }
  k_zero_f32<<<cdiv((size_t)2 * 512, TB), TB, 0, stream>>>(gsums, 1024);
  k_stats<<<statsBlk, 256, 0, stream>>>(p1, 512, gsums);
  k_finalize<<<cdiv(512, TB), TB, 0, stream>>>(gsums, 512, meanb, varb);
  k_bn_apply<<<cdiv((size_t)N_NODESC * 512, TB), TB, 0, stream>>>(
      p1, p_g1, p_be1, meanb, varb, 512, (size_t)N_NODESC * 512);

  { dim3 grid(gemmBx, 256 / 64);
    k_gemm_wmma<<<grid, 128, 0, stream>>>(p1, p_w2, p_b2, nullptr, nullptr, p2, 512, 256, 1); }
  k_zero_f32<<<cdiv((size_t)2 * 256, TB), TB, 0, stream>>>(gsums, 512);
  k_stats<<<statsBlk, 256, 0, stream>>>(p2, 256, gsums);
  k_finalize<<<cdiv(256, TB), TB, 0, stream>>>(gsums, 256, meanb, varb);
  k_bn_apply<<<cdiv((size_t)N_NODESC * 256, TB), TB, 0, stream>>>(
      p2, p_g2, p_be2, meanb, varb, 256, (size_t)N_NODESC * 256);

  { dim3 grid(gemmBx, 128 / 64);
    k_gemm_wmma<<<grid, 128, 0, stream>>>(p2, p_w3, p_b3, nullptr, nullptr, p3, 256, 128, 1); }
  k_zero_f32<<<cdiv((size_t)2 * 128, TB), TB, 0, stream>>>(gsums, 256);
  k_stats<<<statsBlk, 256, 0, stream>>>(p3, 128, gsums);
  k_finalize<<<cdiv(128, TB), TB, 0, stream>>>(gsums, 128, meanb, varb);
  k_bn_apply<<<cdiv((size_t)N_NODESC * 128, TB), TB, 0, stream>>>(
      p3, p_g3, p_be3, meanb, varb, 128, (size_t)N_NODESC * 128);

  // --- output head ---
  k_final<<<nodeBlk, TB, 0, stream>>>(p3, out_w, out_b, outp);
}